// AttentionSumReader_14843406975748
// MI455X (gfx1250) — compile-verified
//
#include <hip/hip_runtime.h>

typedef __attribute__((ext_vector_type(2))) float v2f;
typedef __attribute__((ext_vector_type(8))) float v8f;

#define BATCH        64
#define SEQ          4096
#define EMB          128
#define NUM_ENT      600
#define OUT_ENT      512
#define NTHREADS     256
#define NWAVES       (NTHREADS / 32)
#define TILES_PER_WG 16          // 16 tiles * 16 rows = 256 seq positions per block
#define EPS          1e-9f

// ---------------------------------------------------------------------------
// Kernel 1: masked scores via V_WMMA_F32_16X16X4_F32.
// grid = (B, SEQ/(16*TILES_PER_WG)) = (64, 16); block = 256 (8 waves).
// Each wave owns 16-row score tiles; D(16x16) = A(16x4) x B(4x16) + C,
// where B broadcasts the query k-chunk into every column, so every column
// of the accumulator equals the score vector. Purely bandwidth-bound.
// ---------------------------------------------------------------------------
__launch_bounds__(NTHREADS)
__global__ void asr_scores_wmma(const float* __restrict__ doc_emb,
                                const float* __restrict__ query_emb,
                                const int*   __restrict__ seq_length,
                                float*       __restrict__ scores)
{
    __shared__ float s_q[EMB];

    const int b     = blockIdx.x;
    const int chunk = blockIdx.y;              // which group of 16 tiles
    const int tid   = threadIdx.x;
    const int lane  = tid & 31;
    const int wid   = tid >> 5;
    const int half  = lane >> 4;               // 0: K pair {0,1}-ish, 1: {2,3}-ish
    const int row   = lane & 15;               // A-matrix row within tile

    if (tid < EMB) s_q[tid] = query_emb[b * EMB + tid];
    const int len = max(seq_length[b], 1);
    __syncthreads();

    const int tile0 = chunk * TILES_PER_WG;

    for (int t = wid; t < TILES_PER_WG; t += NWAVES) {
        const int tile = tile0 + t;
        const float* arow =
            doc_emb + ((size_t)b * SEQ + (size_t)tile * 16 + row) * EMB + half * 2;

        v8f c = {0.f, 0.f, 0.f, 0.f, 0.f, 0.f, 0.f, 0.f};
        #pragma unroll
        for (int k = 0; k < EMB / 4; ++k) {
            // A fragment: lane holds doc[row][4k + 2*half .. +1]  (contiguous b64)
            v2f a = *(const v2f*)(arow + 4 * k);
            // B fragment (column-broadcast query): identical index pattern
            v2f q = *(const v2f*)(&s_q[4 * k + half * 2]);
            c = __builtin_amdgcn_wmma_f32_16x16x4_f32(
                    false, a, false, q, (short)0, c, false, false);
        }

        // C layout: VGPR r = row r (lanes 0-15) / row 8+r (lanes 16-31);
        // all columns identical, so lanes 0 and 16 write the 16 scores.
        if (row == 0) {
            const int sbase = tile * 16 + half * 8;
            #pragma unroll
            for (int r = 0; r < 8; ++r) {
                const int s = sbase + r;
                scores[(size_t)b * SEQ + s] = (s < len) ? c[r] : -1e30f;
            }
        }
    }
}

// ---------------------------------------------------------------------------
// Kernel 2: per-batch softmax + entity segment-sum + log.
// grid = (B); block = 256. Traffic is ~2 MB total -> negligible.
// ---------------------------------------------------------------------------
__launch_bounds__(NTHREADS)
__global__ void asr_softmax_segsum(const float* __restrict__ scores,
                                   const int*   __restrict__ doc_ids,
                                   const int*   __restrict__ seq_length,
                                   float*       __restrict__ out)
{
    __shared__ float s_exp[SEQ];
    __shared__ float s_esum[NUM_ENT];
    __shared__ float s_red[NWAVES];
    __shared__ float s_bcast[2];

    const int b    = blockIdx.x;
    const int tid  = threadIdx.x;
    const int lane = tid & 31;
    const int wid  = tid >> 5;
    const int len  = max(seq_length[b], 1);

    for (int e = tid; e < NUM_ENT; e += NTHREADS) s_esum[e] = 0.0f;

    // ---- pass 1: max ----
    float lmax = -3.4e38f;
    for (int s = tid; s < SEQ; s += NTHREADS)
        lmax = fmaxf(lmax, scores[(size_t)b * SEQ + s]);
    #pragma unroll
    for (int off = 16; off > 0; off >>= 1)
        lmax = fmaxf(lmax, __shfl_xor(lmax, off, 32));
    if (lane == 0) s_red[wid] = lmax;
    __syncthreads();
    if (tid == 0) {
        float m = s_red[0];
        for (int w = 1; w < NWAVES; ++w) m = fmaxf(m, s_red[w]);
        s_bcast[0] = m;
    }
    __syncthreads();
    const float smax = s_bcast[0];

    // ---- pass 2: exp + sum (masked rows underflow to exactly 0) ----
    float lsum = 0.0f;
    for (int s = tid; s < SEQ; s += NTHREADS) {
        float e = expf(scores[(size_t)b * SEQ + s] - smax);
        s_exp[s] = e;
        lsum += e;
    }
    #pragma unroll
    for (int off = 16; off > 0; off >>= 1)
        lsum += __shfl_xor(lsum, off, 32);
    if (lane == 0) s_red[wid] = lsum;
    __syncthreads();
    if (tid == 0) {
        float t = 0.0f;
        for (int w = 0; w < NWAVES; ++w) t += s_red[w];
        s_bcast[1] = t;
    }
    __syncthreads();
    const float ssum = s_bcast[1];

    // ---- pass 3: unnormalized segment sum over entity ids (LDS f32 atomics) ----
    for (int s = tid; s < len; s += NTHREADS) {
        const int id = doc_ids[(size_t)b * SEQ + s];
        atomicAdd(&s_esum[id], s_exp[s]);
    }
    __syncthreads();

    // ---- pass 4: logits = log(segsum/sumexp + eps) for entities [0,512) ----
    const float inv = 1.0f / ssum;
    for (int e = tid; e < OUT_ENT; e += NTHREADS)
        out[(size_t)b * OUT_ENT + e] = logf(s_esum[e] * inv + EPS);
}

// ---------------------------------------------------------------------------
extern "C" void kernel_launch(void* const* d_in, const int* in_sizes, int n_in,
                              void* d_out, int out_size, void* d_ws, size_t ws_size,
                              hipStream_t stream)
{
    const float* doc_emb   = (const float*)d_in[0];   // [64,4096,128] f32
    const float* query_emb = (const float*)d_in[1];   // [64,128]      f32
    const int*   doc_ids   = (const int*)  d_in[2];   // [64,4096]     i32
    const int*   seq_len   = (const int*)  d_in[3];   // [64]          i32
    float*       out       = (float*)d_out;           // [64,512]      f32
    float*       scores    = (float*)d_ws;            // [64,4096] f32 = 1 MB scratch

    dim3 grid1(BATCH, SEQ / (16 * TILES_PER_WG));     // (64, 16)
    asr_scores_wmma<<<grid1, NTHREADS, 0, stream>>>(doc_emb, query_emb, seq_len, scores);

    asr_softmax_segsum<<<BATCH, NTHREADS, 0, stream>>>(scores, doc_ids, seq_len, out);
}